// S_ATT_57724360458995
// MI455X (gfx1250) — compile-verified
//
#include <hip/hip_runtime.h>
#include <math.h>

#define D 256
#define LL 512
#define NB 64

typedef __attribute__((ext_vector_type(2))) float v2f;
typedef __attribute__((ext_vector_type(8))) float v8f;

__device__ __forceinline__ v8f wmma_f32(v2f a, v2f b, v8f c) {
  // V_WMMA_F32_16X16X4_F32 : D = A(16x4, f32) * B(4x16, f32) + C(16x16, f32)
  return __builtin_amdgcn_wmma_f32_16x16x4_f32(
      false, a, false, b, (short)0, c, false, false);
}

__device__ __forceinline__ float wave_sum(float v) {
  v += __shfl_xor(v, 16, 32);
  v += __shfl_xor(v, 8, 32);
  v += __shfl_xor(v, 4, 32);
  v += __shfl_xor(v, 2, 32);
  v += __shfl_xor(v, 1, 32);
  return v;
}

__device__ __forceinline__ float wave_max(float v) {
  v = fmaxf(v, __shfl_xor(v, 16, 32));
  v = fmaxf(v, __shfl_xor(v, 8, 32));
  v = fmaxf(v, __shfl_xor(v, 4, 32));
  v = fmaxf(v, __shfl_xor(v, 2, 32));
  v = fmaxf(v, __shfl_xor(v, 1, 32));
  return v;
}

// ---------------- transpose helpers (make B operands K-major) --------------

__global__ void __launch_bounds__(256) k_transpose_w(const float* __restrict__ W,
                                                     float* __restrict__ Wt) {
  const int t = blockIdx.x * 256 + threadIdx.x;   // 65536 elements
  const int o = t >> 8, i = t & 255;              // W[o,i] -> Wt[i,o]
  Wt[i * D + o] = W[t];
}

__global__ void __launch_bounds__(256) k_transpose_x(const float* __restrict__ X,
                                                     float* __restrict__ Xt) {
  __shared__ float tile[32 * 33];
  const int l0 = blockIdx.x * 32;
  const int k0 = blockIdx.y * 32;
  const int b  = blockIdx.z;
  const int tid = threadIdx.x;
  for (int i = tid; i < 1024; i += 256) {
    const int lr = i >> 5, kc = i & 31;
    tile[lr * 33 + kc] = X[((size_t)(b * LL) + l0 + lr) * D + k0 + kc];
  }
  __syncthreads();
  for (int i = tid; i < 1024; i += 256) {
    const int kr = i >> 5, lc = i & 31;
    Xt[((size_t)(b * D) + k0 + kr) * LL + l0 + lc] = tile[lc * 33 + kr];
  }
}

// ---------------- Q projection: Q = relu(X * Wq^T + bq) --------------------
// 64 rows x 32 cols per wave; fused 2 N-tiles x 4 M-tiles per k-step.

__global__ void __launch_bounds__(256) k_qproj(const float* __restrict__ X,
                                               const float* __restrict__ Wt,
                                               const float* __restrict__ bias,
                                               float* __restrict__ Q) {
  __shared__ float Xs[64 * 257];
  const int m0 = blockIdx.x * 64;
  const int tid = threadIdx.x;
  const float4* __restrict__ X4 = (const float4*)X;
  for (int idx = tid; idx < 64 * 64; idx += 256) {   // 64 rows x 64 float4
    const int r = idx >> 6, c4 = (idx & 63) << 2;
    const float4 v = X4[((size_t)(m0 + r) << 6) + (idx & 63)];
    Xs[r * 257 + c4 + 0] = v.x;
    Xs[r * 257 + c4 + 1] = v.y;
    Xs[r * 257 + c4 + 2] = v.z;
    Xs[r * 257 + c4 + 3] = v.w;
  }
  __syncthreads();
  const int wave = tid >> 5, lane = tid & 31;
  const int lrow = lane & 15, lhi = lane >> 4;
  const int ncol0 = (wave * 2) * 16 + lrow;

  v8f acc[2][4] = {};
#pragma unroll 1
  for (int kc = 0; kc < D; kc += 64) {
    if (kc + 64 < D) {
      __builtin_prefetch(&Wt[(kc + 64) * D + ncol0], 0, 1);
      __builtin_prefetch(&Wt[(kc + 64) * D + ncol0 + 16], 0, 1);
    }
#pragma unroll 4
    for (int k0 = kc; k0 < kc + 64; k0 += 4) {
      const int kk = k0 + (lhi << 1);
      v2f a[4];
#pragma unroll
      for (int mt = 0; mt < 4; ++mt) {
        a[mt].x = Xs[(mt * 16 + lrow) * 257 + kk];
        a[mt].y = Xs[(mt * 16 + lrow) * 257 + kk + 1];
      }
#pragma unroll
      for (int nj = 0; nj < 2; ++nj) {
        v2f bb;
        bb.x = Wt[kk * D + ncol0 + nj * 16];
        bb.y = Wt[(kk + 1) * D + ncol0 + nj * 16];
#pragma unroll
        for (int mt = 0; mt < 4; ++mt) acc[nj][mt] = wmma_f32(a[mt], bb, acc[nj][mt]);
      }
    }
  }
#pragma unroll
  for (int nj = 0; nj < 2; ++nj) {
    const int ncol = ncol0 + nj * 16;
    const float bn = bias[ncol];
#pragma unroll
    for (int mt = 0; mt < 4; ++mt)
#pragma unroll
      for (int v = 0; v < 8; ++v) {
        const int m = mt * 16 + v + (lhi << 3);
        const float val = acc[nj][mt][v] + bn;
        Q[(size_t)(m0 + m) * D + ncol] = val > 0.f ? val : 0.f;
      }
  }
}

// ------- attention: scores -> alpha-entmax (bisection) -> attn = A*X -------
// 32 query rows per block.

__global__ void __launch_bounds__(256) k_attn(const float* __restrict__ X,
                                              const float* __restrict__ Xt,
                                              const float* __restrict__ Q,
                                              const int* __restrict__ alphaPtr,
                                              float* __restrict__ attn) {
  __shared__ float Qs[32 * 257];
  __shared__ float Ss[32 * 521];
  const int b  = blockIdx.x >> 4;
  const int q0 = (blockIdx.x & 15) * 32;
  const int tid = threadIdx.x;
  const int wave = tid >> 5, lane = tid & 31;
  const int lrow = lane & 15, lhi = lane >> 4;

  const float4* __restrict__ Q4 = (const float4*)Q;
  for (int idx = tid; idx < 32 * 64; idx += 256) {
    const int r = idx >> 6, c4 = (idx & 63) << 2;
    const float4 v = Q4[(((size_t)b * LL + q0 + r) << 6) + (idx & 63)];
    Qs[r * 257 + c4 + 0] = v.x;
    Qs[r * 257 + c4 + 1] = v.y;
    Qs[r * 257 + c4 + 2] = v.z;
    Qs[r * 257 + c4 + 3] = v.w;
  }
  __syncthreads();

  // scores tile [32 x 512] = Q_tile * X[b]^T / sqrt(D)
  // fused: 4 N-tiles (wave, wave+8, wave+16, wave+24) x 2 M-tiles
  {
    v8f acc[4][2] = {};
    const float* __restrict__ Xtb = Xt + (size_t)b * D * LL;
#pragma unroll 1
    for (int kc = 0; kc < D; kc += 64) {
      if (kc + 64 < D) {
#pragma unroll
        for (int j = 0; j < 4; ++j)
          __builtin_prefetch(&Xtb[(size_t)(kc + 64) * LL + (wave + 8 * j) * 16 + lrow], 0, 1);
      }
#pragma unroll 4
      for (int k0 = kc; k0 < kc + 64; k0 += 4) {
        const int kk = k0 + (lhi << 1);
        v2f a0, a1;
        a0.x = Qs[lrow * 257 + kk];
        a0.y = Qs[lrow * 257 + kk + 1];
        a1.x = Qs[(16 + lrow) * 257 + kk];
        a1.y = Qs[(16 + lrow) * 257 + kk + 1];
#pragma unroll
        for (int j = 0; j < 4; ++j) {
          const int ncol = (wave + 8 * j) * 16 + lrow;
          v2f bb;
          bb.x = Xtb[(size_t)kk * LL + ncol];
          bb.y = Xtb[(size_t)(kk + 1) * LL + ncol];
          acc[j][0] = wmma_f32(a0, bb, acc[j][0]);
          acc[j][1] = wmma_f32(a1, bb, acc[j][1]);
        }
      }
    }
#pragma unroll
    for (int j = 0; j < 4; ++j) {
      const int n0 = (wave + 8 * j) * 16;
#pragma unroll
      for (int mt = 0; mt < 2; ++mt)
#pragma unroll
        for (int v = 0; v < 8; ++v) {
          const int m = mt * 16 + v + (lhi << 3);
          Ss[m * 521 + n0 + lrow] = acc[j][mt][v] * (1.0f / 16.0f);
        }
    }
  }
  __syncthreads();

  // alpha-entmax via 50-step bisection; each wave owns 4 rows, 16 cols/lane
  const float alphaf = (float)(*alphaPtr);
  const float am1 = alphaf - 1.0f;
  const float inv = 1.0f / am1;
  const bool lin = (am1 == 1.0f);   // sparsemax fast path (alpha == 2)
  for (int r = wave * 4; r < wave * 4 + 4; ++r) {
    float xv[16];
    float mx = -3.0e38f;
#pragma unroll
    for (int j = 0; j < 16; ++j) {
      xv[j] = Ss[r * 521 + lane + 32 * j] * am1;
      mx = fmaxf(mx, xv[j]);
    }
    mx = wave_max(mx);
    float tau_lo = mx - 1.0f;
    const float tau_hi = mx - powf(1.0f / (float)LL, am1);
    float dm = tau_hi - tau_lo;
    float tau_m = tau_lo;
    for (int it = 0; it < 50; ++it) {
      dm *= 0.5f;
      tau_m = tau_lo + dm;
      float s = 0.f;
#pragma unroll
      for (int j = 0; j < 16; ++j) {
        const float p = fmaxf(xv[j] - tau_m, 0.f);
        s += lin ? p : powf(p, inv);
      }
      s = wave_sum(s);
      if (s - 1.0f >= 0.f) tau_lo = tau_m;   // wave-uniform branch
    }
    float pm[16];
    float s = 0.f;
#pragma unroll
    for (int j = 0; j < 16; ++j) {
      float p = fmaxf(xv[j] - tau_m, 0.f);
      p = lin ? p : powf(p, inv);
      pm[j] = p;
      s += p;
    }
    s = wave_sum(s);
    const float rs = 1.0f / s;
#pragma unroll
    for (int j = 0; j < 16; ++j) Ss[r * 521 + lane + 32 * j] = pm[j] * rs;
  }
  __syncthreads();

  // attn_tile [32 x 256] = A_tile [32 x 512] * X[b] [512 x 256]
  // fused: 2 N-tiles x 2 M-tiles
  {
    v8f acc[2][2] = {};
    const float* __restrict__ Xb = X + (size_t)b * LL * D;
    const int ncol0 = (wave * 2) * 16 + lrow;
#pragma unroll 1
    for (int kc = 0; kc < LL; kc += 64) {
      if (kc + 64 < LL) {
        __builtin_prefetch(&Xb[(size_t)(kc + 64) * D + ncol0], 0, 1);
        __builtin_prefetch(&Xb[(size_t)(kc + 64) * D + ncol0 + 16], 0, 1);
      }
#pragma unroll 4
      for (int k0 = kc; k0 < kc + 64; k0 += 4) {
        const int kk = k0 + (lhi << 1);
        v2f a0, a1;
        a0.x = Ss[lrow * 521 + kk];
        a0.y = Ss[lrow * 521 + kk + 1];
        a1.x = Ss[(16 + lrow) * 521 + kk];
        a1.y = Ss[(16 + lrow) * 521 + kk + 1];
#pragma unroll
        for (int nj = 0; nj < 2; ++nj) {
          v2f bb;
          bb.x = Xb[(size_t)kk * D + ncol0 + nj * 16];
          bb.y = Xb[(size_t)(kk + 1) * D + ncol0 + nj * 16];
          acc[nj][0] = wmma_f32(a0, bb, acc[nj][0]);
          acc[nj][1] = wmma_f32(a1, bb, acc[nj][1]);
        }
      }
    }
#pragma unroll
    for (int nj = 0; nj < 2; ++nj)
#pragma unroll
      for (int mt = 0; mt < 2; ++mt)
#pragma unroll
        for (int v = 0; v < 8; ++v) {
          const int m = mt * 16 + v + (lhi << 3);
          attn[((size_t)b * LL + q0 + m) * D + ncol0 + nj * 16] = acc[nj][mt][v];
        }
  }
}

// ------- FFN + residual + LayerNorm + split-halves store -------------------
// 32 rows per block; fused 2 N-tiles x 2 M-tiles per GEMM.

__global__ void __launch_bounds__(256) k_ffn(const float* __restrict__ attn,
                                             const float* __restrict__ W1t,
                                             const float* __restrict__ b1,
                                             const float* __restrict__ W2t,
                                             const float* __restrict__ b2,
                                             const float* __restrict__ gamma,
                                             const float* __restrict__ beta,
                                             float* __restrict__ out) {
  __shared__ float As[32 * 257];
  __shared__ float Hs[32 * 257];
  __shared__ float Fs[32 * 257];
  const int g0 = blockIdx.x * 32;
  const int tid = threadIdx.x;
  const int wave = tid >> 5, lane = tid & 31;
  const int lrow = lane & 15, lhi = lane >> 4;
  const int ncol0 = (wave * 2) * 16 + lrow;

  const float4* __restrict__ A4 = (const float4*)attn;
  for (int idx = tid; idx < 32 * 64; idx += 256) {
    const int r = idx >> 6, c4 = (idx & 63) << 2;
    const float4 v = A4[((size_t)(g0 + r) << 6) + (idx & 63)];
    As[r * 257 + c4 + 0] = v.x;
    As[r * 257 + c4 + 1] = v.y;
    As[r * 257 + c4 + 2] = v.z;
    As[r * 257 + c4 + 3] = v.w;
  }
  __syncthreads();

  // h1 = relu(attn * W1^T + b1)
  {
    v8f acc[2][2] = {};
#pragma unroll 1
    for (int kc = 0; kc < D; kc += 64) {
      if (kc + 64 < D) {
        __builtin_prefetch(&W1t[(kc + 64) * D + ncol0], 0, 1);
        __builtin_prefetch(&W1t[(kc + 64) * D + ncol0 + 16], 0, 1);
      }
#pragma unroll 4
      for (int k0 = kc; k0 < kc + 64; k0 += 4) {
        const int kk = k0 + (lhi << 1);
        v2f a0, a1;
        a0.x = As[lrow * 257 + kk];
        a0.y = As[lrow * 257 + kk + 1];
        a1.x = As[(16 + lrow) * 257 + kk];
        a1.y = As[(16 + lrow) * 257 + kk + 1];
#pragma unroll
        for (int nj = 0; nj < 2; ++nj) {
          v2f bb;
          bb.x = W1t[kk * D + ncol0 + nj * 16];
          bb.y = W1t[(kk + 1) * D + ncol0 + nj * 16];
          acc[nj][0] = wmma_f32(a0, bb, acc[nj][0]);
          acc[nj][1] = wmma_f32(a1, bb, acc[nj][1]);
        }
      }
    }
#pragma unroll
    for (int nj = 0; nj < 2; ++nj) {
      const int ncol = ncol0 + nj * 16;
      const float bn = b1[ncol];
#pragma unroll
      for (int mt = 0; mt < 2; ++mt)
#pragma unroll
        for (int v = 0; v < 8; ++v) {
          const int m = mt * 16 + v + (lhi << 3);
          const float val = acc[nj][mt][v] + bn;
          Hs[m * 257 + ncol] = val > 0.f ? val : 0.f;
        }
    }
  }
  __syncthreads();

  // ffn = h1 * W2^T + b2 ; Fs = ffn + attn (residual)
  {
    v8f acc[2][2] = {};
#pragma unroll 1
    for (int kc = 0; kc < D; kc += 64) {
      if (kc + 64 < D) {
        __builtin_prefetch(&W2t[(kc + 64) * D + ncol0], 0, 1);
        __builtin_prefetch(&W2t[(kc + 64) * D + ncol0 + 16], 0, 1);
      }
#pragma unroll 4
      for (int k0 = kc; k0 < kc + 64; k0 += 4) {
        const int kk = k0 + (lhi << 1);
        v2f a0, a1;
        a0.x = Hs[lrow * 257 + kk];
        a0.y = Hs[lrow * 257 + kk + 1];
        a1.x = Hs[(16 + lrow) * 257 + kk];
        a1.y = Hs[(16 + lrow) * 257 + kk + 1];
#pragma unroll
        for (int nj = 0; nj < 2; ++nj) {
          v2f bb;
          bb.x = W2t[kk * D + ncol0 + nj * 16];
          bb.y = W2t[(kk + 1) * D + ncol0 + nj * 16];
          acc[nj][0] = wmma_f32(a0, bb, acc[nj][0]);
          acc[nj][1] = wmma_f32(a1, bb, acc[nj][1]);
        }
      }
    }
#pragma unroll
    for (int nj = 0; nj < 2; ++nj) {
      const int ncol = ncol0 + nj * 16;
      const float bn = b2[ncol];
#pragma unroll
      for (int mt = 0; mt < 2; ++mt)
#pragma unroll
        for (int v = 0; v < 8; ++v) {
          const int m = mt * 16 + v + (lhi << 3);
          Fs[m * 257 + ncol] = acc[nj][mt][v] + bn + As[m * 257 + ncol];
        }
    }
  }
  __syncthreads();

  // LayerNorm per row (4 rows per wave); split-halves output layout
  for (int r = wave * 4; r < wave * 4 + 4; ++r) {
    float x[8];
    float s = 0.f;
#pragma unroll
    for (int j = 0; j < 8; ++j) {
      x[j] = Fs[r * 257 + lane + 32 * j];
      s += x[j];
    }
    s = wave_sum(s);
    const float mu = s * (1.0f / 256.0f);
    float vs = 0.f;
#pragma unroll
    for (int j = 0; j < 8; ++j) {
      const float d = x[j] - mu;
      vs += d * d;
    }
    vs = wave_sum(vs);
    const float rstd = rsqrtf(vs * (1.0f / 256.0f) + 1e-12f);
    const int g = g0 + r;
    const int bidx = g >> 9;          // batch
    const int l = g & 511;            // sequence pos
    const size_t base = (l < 256)
        ? (size_t)(bidx * 256 + l) * D
        : (size_t)(NB * 256) * D + (size_t)(bidx * 256 + (l - 256)) * D;
#pragma unroll
    for (int j = 0; j < 8; ++j) {
      const int col = lane + 32 * j;
      out[base + col] = (x[j] - mu) * rstd * gamma[col] + beta[col];
    }
  }
}

// ---------------------------------------------------------------------------

extern "C" void kernel_launch(void* const* d_in, const int* in_sizes, int n_in,
                              void* d_out, int out_size, void* d_ws, size_t ws_size,
                              hipStream_t stream) {
  (void)in_sizes; (void)n_in; (void)out_size; (void)ws_size;
  const float* X  = (const float*)d_in[0];
  const int*   al = (const int*)d_in[1];
  const float* Wq = (const float*)d_in[2];
  const float* bq = (const float*)d_in[3];
  const float* W1 = (const float*)d_in[4];
  const float* b1 = (const float*)d_in[5];
  const float* W2 = (const float*)d_in[6];
  const float* b2 = (const float*)d_in[7];
  const float* gm = (const float*)d_in[8];
  const float* bt = (const float*)d_in[9];
  float* out = (float*)d_out;

  float* ws  = (float*)d_ws;
  float* Wqt = ws;                    // 256*256
  float* W1t = Wqt + 65536;           // 256*256
  float* W2t = W1t + 65536;           // 256*256
  float* Xt  = W2t + 65536;           // 64*256*512
  float* Qw  = Xt + (size_t)NB * D * LL;   // 64*512*256
  float* Aw  = Qw + (size_t)NB * LL * D;   // 64*512*256

  k_transpose_w<<<256, 256, 0, stream>>>(Wq, Wqt);
  k_transpose_w<<<256, 256, 0, stream>>>(W1, W1t);
  k_transpose_w<<<256, 256, 0, stream>>>(W2, W2t);
  k_transpose_x<<<dim3(16, 8, NB), 256, 0, stream>>>(X, Xt);
  k_qproj<<<(NB * LL) / 64, 256, 0, stream>>>(X, Wqt, bq, Qw);
  k_attn<<<NB * (LL / 32), 256, 0, stream>>>(X, Xt, Qw, al, Aw);
  k_ffn<<<(NB * LL) / 32, 256, 0, stream>>>(Aw, W1t, b1, W2t, b2, gm, bt, out);
}